// GNN_prelu_edge_50689204027575
// MI455X (gfx1250) — compile-verified
//
#include <hip/hip_runtime.h>

// ---------------------------------------------------------------------------
// Hetero-GNN (SAGE + edge attrs) forward for MI455X / gfx1250 (wave32, WMMA).
// Phase 1: per-edge scatter-mean accumulation (L2-resident f32 atomics).
// Phase 2: per-16-node-tile fused SAGE epilogue using V_WMMA_F32_16X16X4_F32.
// ---------------------------------------------------------------------------

typedef float v2f __attribute__((ext_vector_type(2)));
typedef float v8f __attribute__((ext_vector_type(8)));

#define N_PFAS 50000
#define N_GW   200000
#define N_SW   20000
#define D      32

// ------------------------------ utility ------------------------------------

__global__ void zero_f32_kernel(float* __restrict__ p, size_t n) {
    size_t i      = (size_t)blockIdx.x * blockDim.x + threadIdx.x;
    size_t stride = (size_t)gridDim.x * blockDim.x;
    for (; i < n; i += stride) p[i] = 0.0f;
}

// ------------------------- edge scatter kernels -----------------------------
// One wave32 per edge; lane l owns feature l. Gather is a coalesced 128B row
// read; accumulation is 32 coalesced global_atomic_add_f32 into the L2-resident
// accumulator. Lane 0 counts the edge for the mean divisor.

__global__ void __launch_bounds__(256)
edge_scatter_mean(const float* __restrict__ xsrc,
                  const int*   __restrict__ src,
                  const int*   __restrict__ dst,
                  float*       __restrict__ acc,
                  float*       __restrict__ cnt,
                  int E)
{
    int e    = (int)(((size_t)blockIdx.x * blockDim.x + threadIdx.x) >> 5);
    int lane = threadIdx.x & 31;
    if (e >= E) return;
    int s = src[e];
    int d = dst[e];
    float v = xsrc[(size_t)s * D + lane];
    atomicAdd(&acc[(size_t)d * D + lane], v);
    if (lane == 0) atomicAdd(&cnt[d], 1.0f);
}

// edge_attr scatter-"set": val = ea[e] @ We + be, overwrite at dst (racing
// store matches the reference's arbitrary-winner scatter_ semantics).
__global__ void __launch_bounds__(256)
edge_attr_set(const float* __restrict__ ea,    // [E,3]
              const int*   __restrict__ dst,
              const float* __restrict__ We,    // [3,32]
              const float* __restrict__ be,    // [32]
              float*       __restrict__ eacc,  // [n_dst,32]
              int E)
{
    int e    = (int)(((size_t)blockIdx.x * blockDim.x + threadIdx.x) >> 5);
    int lane = threadIdx.x & 31;
    if (e >= E) return;
    int d    = dst[e];
    float e0 = ea[(size_t)e * 3 + 0];
    float e1 = ea[(size_t)e * 3 + 1];
    float e2 = ea[(size_t)e * 3 + 2];
    float v  = fmaf(e0, We[0 * D + lane],
               fmaf(e1, We[1 * D + lane],
               fmaf(e2, We[2 * D + lane], be[lane])));
    eacc[(size_t)d * D + lane] = v;
}

// --------------------------- node WMMA kernel -------------------------------
// C(16x16,f32) += A(16x4,f32) x B(4x16,f32) per V_WMMA_F32_16X16X4_F32.
// ISA layouts (7.12.2):
//   A: lane -> M = lane&15, VGPR v -> K = 4*kc + 2*(lane>>4) + v
//   B: lane -> N = lane&15, VGPR v -> K = 4*kc + 2*(lane>>4) + v
//   C/D: lane -> N = lane&15, VGPR v -> M = v + 8*(lane>>4)

__device__ __forceinline__ void gemm_acc_32(const float* __restrict__ Arows, // [n,32]
                                            float scale,
                                            const float* __restrict__ W,    // [32,32]
                                            int m, int lg, int ln,
                                            v8f& c0, v8f& c1)
{
#pragma unroll
    for (int kc = 0; kc < 8; ++kc) {
        const int k0 = kc * 4 + 2 * lg;
        v2f a;
        a.x = Arows[(size_t)m * D + k0]     * scale;
        a.y = Arows[(size_t)m * D + k0 + 1] * scale;
        v2f b0, b1;
        b0.x = W[(size_t)k0 * D + ln];
        b0.y = W[(size_t)(k0 + 1) * D + ln];
        b1.x = W[(size_t)k0 * D + 16 + ln];
        b1.y = W[(size_t)(k0 + 1) * D + 16 + ln];
        c0 = __builtin_amdgcn_wmma_f32_16x16x4_f32(false, a, false, b0,
                                                   (short)0, c0, false, false);
        c1 = __builtin_amdgcn_wmma_f32_16x16x4_f32(false, a, false, b1,
                                                   (short)0, c1, false, false);
    }
}

// One wave per 16-node tile; 4 waves (128 threads) per block.
// out is [n,32] (h) unless Wout != nullptr, in which case out is [n] (y with
// fused final linear + PReLU).
__global__ void __launch_bounds__(128)
sage_node_wmma(float*       __restrict__ out,
               int n_nodes,
               const float* __restrict__ xdst,
               const float* __restrict__ acc1, const float* __restrict__ cnt1,
               const float* __restrict__ Wl1,  const float* __restrict__ bl1,
               const float* __restrict__ acc2, const float* __restrict__ cnt2,
               const float* __restrict__ Wl2,  const float* __restrict__ bl2,
               const float* __restrict__ Wr1,
               const float* __restrict__ Wr2,
               const float* __restrict__ eacc,
               const float* __restrict__ Wout,
               const float* __restrict__ bout,
               const float* __restrict__ aprelu)
{
    const int lane = threadIdx.x & 31;
    const int wave = threadIdx.x >> 5;
    const int tile = blockIdx.x * 4 + wave;
    const int row0 = tile * 16;
    if (row0 >= n_nodes) return;   // wave-uniform exit: EXEC stays all-ones

    const int lg = lane >> 4;
    const int ln = lane & 15;
    const int m  = row0 + ln;      // A-row owned by this lane

    const float inv1 = 1.0f / fmaxf(cnt1[m], 1.0f);

    v8f c0 = {};
    v8f c1 = {};

    gemm_acc_32(acc1, inv1, Wl1, m, lg, ln, c0, c1);      // mean1 @ Wl1
    if (acc2) {
        const float inv2 = 1.0f / fmaxf(cnt2[m], 1.0f);
        gemm_acc_32(acc2, inv2, Wl2, m, lg, ln, c0, c1);  // mean2 @ Wl2
    }
    gemm_acc_32(xdst, 1.0f, Wr1, m, lg, ln, c0, c1);      // x_dst @ Wr1
    if (Wr2) gemm_acc_32(xdst, 1.0f, Wr2, m, lg, ln, c0, c1);

    float biasN0 = bl1[ln]      + (bl2 ? bl2[ln]      : 0.0f);
    float biasN1 = bl1[16 + ln] + (bl2 ? bl2[16 + ln] : 0.0f);

    const float w0 = Wout ? Wout[ln]      : 0.0f;
    const float w1 = Wout ? Wout[16 + ln] : 0.0f;

#pragma unroll
    for (int v = 0; v < 8; ++v) {
        const int row = row0 + v + 8 * lg;
        float h0 = c0[v] + biasN0;
        float h1 = c1[v] + biasN1;
        if (eacc) {
            h0 += eacc[(size_t)row * D + ln];
            h1 += eacc[(size_t)row * D + 16 + ln];
        }
        h0 = fmaxf(h0, 0.0f);   // HeteroConv sum + relu
        h1 = fmaxf(h1, 0.0f);
        if (!Wout) {
            out[(size_t)row * D + ln]      = h0;
            out[(size_t)row * D + 16 + ln] = h1;
        } else {
            // fused y = h @ W_out + b_out, PReLU; reduce over the 16 lanes of
            // this lane-group (xor<16 never crosses the group boundary).
            float p = h0 * w0 + h1 * w1;
            p += __shfl_xor(p, 8, 32);
            p += __shfl_xor(p, 4, 32);
            p += __shfl_xor(p, 2, 32);
            p += __shfl_xor(p, 1, 32);
            if (ln == 0) {
                float y = p + bout[0];
                out[row] = (y >= 0.0f) ? y : aprelu[0] * y;
            }
        }
    }
}

// ------------------------------- launcher -----------------------------------

extern "C" void kernel_launch(void* const* d_in, const int* in_sizes, int n_in,
                              void* d_out, int out_size, void* d_ws, size_t ws_size,
                              hipStream_t stream)
{
    (void)in_sizes; (void)n_in; (void)out_size; (void)ws_size;

    const float* x_pfas = (const float*)d_in[0];
    const float* x_gw   = (const float*)d_in[1];
    const float* x_sw   = (const float*)d_in[2];
    const int*   src_pg = (const int*)d_in[3];
    const int*   dst_pg = (const int*)d_in[4];
    const float* ea_pg  = (const float*)d_in[5];
    const int*   src_gp = (const int*)d_in[6];
    const int*   dst_gp = (const int*)d_in[7];
    const float* ea_gp  = (const float*)d_in[8];
    const int*   src_ps = (const int*)d_in[9];
    const int*   dst_ps = (const int*)d_in[10];
    const int*   src_sp = (const int*)d_in[11];
    const int*   dst_sp = (const int*)d_in[12];
    const float* Wl_pg  = (const float*)d_in[13];
    const float* bl_pg  = (const float*)d_in[14];
    const float* Wr_pg  = (const float*)d_in[15];
    const float* We_pg  = (const float*)d_in[16];
    const float* be_pg  = (const float*)d_in[17];
    const float* Wl_gp  = (const float*)d_in[18];
    const float* bl_gp  = (const float*)d_in[19];
    const float* Wr_gp  = (const float*)d_in[20];
    const float* We_gp  = (const float*)d_in[21];
    const float* be_gp  = (const float*)d_in[22];
    const float* Wl_ps  = (const float*)d_in[23];
    const float* bl_ps  = (const float*)d_in[24];
    const float* Wr_ps  = (const float*)d_in[25];
    const float* Wl_sp  = (const float*)d_in[26];
    const float* bl_sp  = (const float*)d_in[27];
    const float* Wr_sp  = (const float*)d_in[28];
    const float* W_out  = (const float*)d_in[29];
    const float* b_out  = (const float*)d_in[30];
    const float* a_prelu= (const float*)d_in[31];

    const int E_PG = 2000000, E_GP = 2000000, E_PS = 1000000, E_SP = 1000000;

    // ---- workspace layout (floats) ----
    float* ws = (float*)d_ws;
    size_t o = 0;
    float* acc_pg  = ws + o; o += (size_t)N_GW   * D;   // pg: mean accum @ gw
    float* cnt_pg  = ws + o; o += (size_t)N_GW;
    float* acc_gp  = ws + o; o += (size_t)N_PFAS * D;   // gp: mean accum @ pfas
    float* cnt_gp  = ws + o; o += (size_t)N_PFAS;
    float* acc_sp  = ws + o; o += (size_t)N_PFAS * D;   // sp: mean accum @ pfas
    float* cnt_sp  = ws + o; o += (size_t)N_PFAS;
    float* acc_ps  = ws + o; o += (size_t)N_SW   * D;   // ps: mean accum @ sw
    float* cnt_ps  = ws + o; o += (size_t)N_SW;
    float* eacc_pg = ws + o; o += (size_t)N_GW   * D;   // edge-attr set @ gw
    float* eacc_gp = ws + o; o += (size_t)N_PFAS * D;   // edge-attr set @ pfas
    const size_t zero_n = o;                            // ~18.56M floats (~74MB)

    float* h_pfas = (float*)d_out;                      // [50000,32]
    float* y_gw   = (float*)d_out + (size_t)N_PFAS * D; // [200000]
    float* h_sw   = y_gw + (size_t)N_GW;                // [20000,32]

    // ---- phase 0: zero accumulators ----
    zero_f32_kernel<<<4096, 256, 0, stream>>>(ws, zero_n);

    // ---- phase 1: edge scatter (one wave32 per edge) ----
    const int TPB = 256, WPB = TPB / 32;
    edge_scatter_mean<<<(E_PG + WPB - 1) / WPB, TPB, 0, stream>>>(
        x_pfas, src_pg, dst_pg, acc_pg, cnt_pg, E_PG);
    edge_scatter_mean<<<(E_GP + WPB - 1) / WPB, TPB, 0, stream>>>(
        x_gw,   src_gp, dst_gp, acc_gp, cnt_gp, E_GP);
    edge_scatter_mean<<<(E_SP + WPB - 1) / WPB, TPB, 0, stream>>>(
        x_sw,   src_sp, dst_sp, acc_sp, cnt_sp, E_SP);
    edge_scatter_mean<<<(E_PS + WPB - 1) / WPB, TPB, 0, stream>>>(
        x_pfas, src_ps, dst_ps, acc_ps, cnt_ps, E_PS);
    edge_attr_set<<<(E_PG + WPB - 1) / WPB, TPB, 0, stream>>>(
        ea_pg, dst_pg, We_pg, be_pg, eacc_pg, E_PG);
    edge_attr_set<<<(E_GP + WPB - 1) / WPB, TPB, 0, stream>>>(
        ea_gp, dst_gp, We_gp, be_gp, eacc_gp, E_GP);

    // ---- phase 2: fused node-level SAGE epilogues (WMMA f32 16x16x4) ----
    const int NODES_PER_BLOCK = 64; // 4 waves x 16-row tiles
    // gw: relu(mean_pg@Wl_pg + bl_pg + x_gw@Wr_pg + eacc_pg) @ W_out + b_out, PReLU
    sage_node_wmma<<<(N_GW + NODES_PER_BLOCK - 1) / NODES_PER_BLOCK, 128, 0, stream>>>(
        y_gw, N_GW, x_gw,
        acc_pg, cnt_pg, Wl_pg, bl_pg,
        nullptr, nullptr, nullptr, nullptr,
        Wr_pg, nullptr,
        eacc_pg,
        W_out, b_out, a_prelu);
    // pfas: relu(mean_gp@Wl_gp + bl_gp + x@Wr_gp + eacc_gp + mean_sp@Wl_sp + bl_sp + x@Wr_sp)
    sage_node_wmma<<<(N_PFAS + NODES_PER_BLOCK - 1) / NODES_PER_BLOCK, 128, 0, stream>>>(
        h_pfas, N_PFAS, x_pfas,
        acc_gp, cnt_gp, Wl_gp, bl_gp,
        acc_sp, cnt_sp, Wl_sp, bl_sp,
        Wr_gp, Wr_sp,
        eacc_gp,
        nullptr, nullptr, nullptr);
    // sw: relu(mean_ps@Wl_ps + bl_ps + x_sw@Wr_ps)
    sage_node_wmma<<<(N_SW + NODES_PER_BLOCK - 1) / NODES_PER_BLOCK, 128, 0, stream>>>(
        h_sw, N_SW, x_sw,
        acc_ps, cnt_ps, Wl_ps, bl_ps,
        nullptr, nullptr, nullptr, nullptr,
        Wr_ps, nullptr,
        nullptr,
        nullptr, nullptr, nullptr);
}